// FlashFFTConvInterface_40492951667434
// MI455X (gfx1250) — compile-verified
//
#include <hip/hip_runtime.h>
#include <math.h>

// CDNA5 / gfx1250: wave32, WMMA f32 16x16x4, 320KB LDS per WGP.
typedef __attribute__((ext_vector_type(2))) float v2f;
typedef __attribute__((ext_vector_type(8))) float v8f;

#define PI_F 3.14159265358979323846f

#define IH 224
#define IW 224
#define NP 256        // padded FFT size
#define NHW 129       // rfft half-spectrum width
#define NC 96
#define NBATCH 8

#define TS 130        // tile row stride (complex elems), pad vs bank conflicts
#define FS 17         // 16x16 LDS matrix stride (bank-conflict pad)

// ---- workspace layout (float offsets) ----
#define WS_F_R   0        // 256 floats : Re(DFT16) row-major, [j][k] = cos(-2pi jk/16)
#define WS_F_I   256      // 256 floats : Im(DFT16)
#define WS_TW    512      // 512 floats : float2 T[c][b] = exp(-2pi i bc/256), idx=c*16+b
#define WS_KHAT  1024     // float2[96][256][129] = fft2 half-spectrum of padded k

// ---- LDS layout (float offsets), dynamic shared ----
#define L_TILE   0                     // 256*130*2 = 66560 floats (complex interleaved)
#define L_SCR    66560                 // per-wave scratch: 272 R + 272 I = 544, x8 waves
#define LDS_FLOATS (66560 + 8 * 544)   // 70912 floats = 283648 bytes (< 320KB WGP LDS)

// ---- optional async global->LDS path (ASYNCcnt) ----
#if __has_builtin(__builtin_amdgcn_global_load_async_to_lds_b32) && __has_builtin(__builtin_amdgcn_s_wait_asynccnt)
#define HAVE_ASYNC_LDS 1
typedef __attribute__((address_space(1))) int as1_int;   // builtin takes AS1 int*
typedef __attribute__((address_space(3))) int as3_int;   // and AS3 int*
#else
#define HAVE_ASYNC_LDS 0
#endif

// ---------------- fragment loaders (WMMA f32 16x16x4 layouts) ----------------
// A (16x4): lanes 0-15 row M=lane, K={4t+0,4t+1}; lanes 16-31: K={4t+2,4t+3}
__device__ __forceinline__ v2f ldfragA(const float* m, int t, int lane) {
  int row = lane & 15;
  int kk  = t * 4 + ((lane >> 4) << 1);
  v2f r;
  r[0] = m[row * FS + kk];
  r[1] = m[row * FS + kk + 1];
  return r;
}
// B (4x16): lanes 0-15 col N=lane, K={4t+0,4t+1}; lanes 16-31: K={4t+2,4t+3}
__device__ __forceinline__ v2f ldfragB(const float* m, int t, int lane) {
  int col = lane & 15;
  int kk  = t * 4 + ((lane >> 4) << 1);
  v2f r;
  r[0] = m[kk * FS + col];
  r[1] = m[(kk + 1) * FS + col];
  return r;
}

// DFT16 fragments held in VGPRs. F16 is symmetric, so the per-lane A-layout
// fragment values equal the B-layout fragment values -> one copy serves both.
struct Ffrag { v2f r[4]; v2f i[4]; };

__device__ __forceinline__ void load_ffrag(const float* ws, int lane, Ffrag& f) {
  int row = lane & 15;
#pragma unroll
  for (int t = 0; t < 4; ++t) {
    int kk = t * 4 + ((lane >> 4) << 1);
    f.r[t][0] = ws[WS_F_R + row * 16 + kk];
    f.r[t][1] = ws[WS_F_R + row * 16 + kk + 1];
    f.i[t][0] = ws[WS_F_I + row * 16 + kk];
    f.i[t][1] = ws[WS_F_I + row * 16 + kk + 1];
  }
}

// G = F * M  (A = F in regs, B = M in LDS): 16 v_wmma, 8 DS fragment loads
__device__ __forceinline__ void cmul_FxM(const Ffrag& f, const float* bR, const float* bI,
                                         int lane, v8f& outR, v8f& outI) {
  v8f pp = {}; v8f qq = {}; v8f ss = {}; v8f tt = {};
#pragma unroll
  for (int t = 0; t < 4; ++t) {
    v2f br = ldfragB(bR, t, lane);
    v2f bi = ldfragB(bI, t, lane);
    pp = __builtin_amdgcn_wmma_f32_16x16x4_f32(false, f.r[t], false, br, (short)0, pp, false, false);
    qq = __builtin_amdgcn_wmma_f32_16x16x4_f32(false, f.i[t], false, bi, (short)0, qq, false, false);
    ss = __builtin_amdgcn_wmma_f32_16x16x4_f32(false, f.r[t], false, bi, (short)0, ss, false, false);
    tt = __builtin_amdgcn_wmma_f32_16x16x4_f32(false, f.i[t], false, br, (short)0, tt, false, false);
  }
  outR = pp - qq;
  outI = ss + tt;
}

// R = H * F  (A = H in LDS, B = F in regs; F symmetric so same fragments)
__device__ __forceinline__ void cmul_MxF(const float* aR, const float* aI, const Ffrag& f,
                                         int lane, v8f& outR, v8f& outI) {
  v8f pp = {}; v8f qq = {}; v8f ss = {}; v8f tt = {};
#pragma unroll
  for (int t = 0; t < 4; ++t) {
    v2f ar = ldfragA(aR, t, lane);
    v2f ai = ldfragA(aI, t, lane);
    pp = __builtin_amdgcn_wmma_f32_16x16x4_f32(false, ar, false, f.r[t], (short)0, pp, false, false);
    qq = __builtin_amdgcn_wmma_f32_16x16x4_f32(false, ai, false, f.i[t], (short)0, qq, false, false);
    ss = __builtin_amdgcn_wmma_f32_16x16x4_f32(false, ar, false, f.i[t], (short)0, ss, false, false);
    tt = __builtin_amdgcn_wmma_f32_16x16x4_f32(false, ai, false, f.r[t], (short)0, tt, false, false);
  }
  outR = pp - qq;
  outI = ss + tt;
}

// Forward 256-point DFT of one sequence per wave. M[a][b] = x[16a+b] in scr.
// Four-step: G = F16*M ; H = T(.)G ; R = H*F16 ; X[c+16d] = R[c][d].
// Inverse is done by callers via idft(x) = conj(dft(conj(x)))/N.
// Result C/D layout: lane -> d = lane&15, vgpr j -> c = j + 8*(lane>>4).
__device__ __forceinline__ void fft256_wave(float* scrR, float* scrI, const Ffrag& f,
                                            v8f twr, v8f twi, int lane,
                                            v8f& outR, v8f& outI) {
  v8f gr, gi;
  cmul_FxM(f, scrR, scrI, lane, gr, gi);
  v8f hr = gr * twr - gi * twi;
  v8f hi = gr * twi + gi * twr;
  int col = lane & 15, rbase = (lane >> 4) * 8;
#pragma unroll
  for (int j = 0; j < 8; ++j) {           // spill H (same-wave DS stays in order)
    scrR[(rbase + j) * FS + col] = hr[j];
    scrI[(rbase + j) * FS + col] = hi[j];
  }
  cmul_MxF(scrR, scrI, f, lane, outR, outI);
}

// ---------------- kernel 1: DFT16 + twiddle tables ----------------
__global__ void __launch_bounds__(256) fftconv_tables_kernel(float* ws) {
  int t = threadIdx.x;          // 256 threads
  int j = t >> 4, k = t & 15;
  float a16 = -2.0f * PI_F * (float)(j * k) / 16.0f;
  ws[WS_F_R + t] = cosf(a16);
  ws[WS_F_I + t] = sinf(a16);
  float a256 = -2.0f * PI_F * (float)(j * k) / 256.0f;  // T[c=j][b=k]
  ws[WS_TW + 2 * t]     = cosf(a256);
  ws[WS_TW + 2 * t + 1] = sinf(a256);
}

__device__ __forceinline__ void load_twiddles(const float* ws, int lane, v8f& twr, v8f& twi) {
  int col = lane & 15, rbase = (lane >> 4) * 8;
#pragma unroll
  for (int j = 0; j < 8; ++j) {
    int idx = (rbase + j) * 16 + col;
    twr[j] = ws[WS_TW + 2 * idx];
    twi[j] = ws[WS_TW + 2 * idx + 1];
  }
}

// Forward half-transform of one padded real 224x224 image into the LDS tile:
// rows 0..223 row-rFFT (keep w<=128), rows 224..255 zero, then 129 column FFTs.
__device__ __forceinline__ void forward_tile(const float* __restrict__ img,
                                             float* tile, float* scrR, float* scrI,
                                             const Ffrag& f, v8f twr, v8f twi,
                                             int tid, int lane, int wave) {
  for (int i = tid; i < NP * TS * 2; i += 256) tile[i] = 0.f;   // zero (covers pad rows)
  __syncthreads();

  for (int r = wave; r < IH; r += 8) {                          // uniform per wave
    const float* x = img + (size_t)r * IW;
#if HAVE_ASYNC_LDS
#pragma unroll
    for (int e = 0; e < 8; ++e) {                               // imag plane = 0
      int n = lane * 8 + e;
      scrI[(n >> 4) * FS + (n & 15)] = 0.f;
    }
    if (lane < (IW / 8)) {                                      // lanes 0..27 carry data
#pragma unroll
      for (int e = 0; e < 8; ++e) {
        int n = lane * 8 + e;
        // per-lane async copy global -> per-wave LDS scratch (ASYNCcnt)
        __builtin_amdgcn_global_load_async_to_lds_b32(
            (as1_int*)(unsigned long long)(x + n),
            (as3_int*)(unsigned)(unsigned long long)(scrR + (n >> 4) * FS + (n & 15)),
            0, 0);
      }
    } else {                                                    // zero pad tail
#pragma unroll
      for (int e = 0; e < 8; ++e) {
        int n = lane * 8 + e;
        scrR[(n >> 4) * FS + (n & 15)] = 0.f;
      }
    }
    __builtin_amdgcn_s_wait_asynccnt(0);
#else
#pragma unroll
    for (int e = 0; e < 8; ++e) {
      int n = lane * 8 + e;
      float v = (n < IW) ? x[n] : 0.f;
      scrR[(n >> 4) * FS + (n & 15)] = v;
      scrI[(n >> 4) * FS + (n & 15)] = 0.f;
    }
#endif
    v8f rr, ri;
    fft256_wave(scrR, scrI, f, twr, twi, lane, rr, ri);
    int col = lane & 15, rbase = (lane >> 4) * 8;
#pragma unroll
    for (int j = 0; j < 8; ++j) {
      int k = (rbase + j) + 16 * col;
      if (k < NHW) {
        tile[(r * TS + k) * 2]     = rr[j];
        tile[(r * TS + k) * 2 + 1] = ri[j];
      }
    }
  }
  __syncthreads();

  for (int q = wave; q < NHW; q += 8) {                         // 129 half-spectrum columns
#pragma unroll
    for (int e = 0; e < 8; ++e) {
      int n = lane * 8 + e;
      scrR[(n >> 4) * FS + (n & 15)] = tile[(n * TS + q) * 2];
      scrI[(n >> 4) * FS + (n & 15)] = tile[(n * TS + q) * 2 + 1];
    }
    v8f rr, ri;
    fft256_wave(scrR, scrI, f, twr, twi, lane, rr, ri);
    int col = lane & 15, rbase = (lane >> 4) * 8;
#pragma unroll
    for (int j = 0; j < 8; ++j) {
      int k = (rbase + j) + 16 * col;
      tile[(k * TS + q) * 2]     = rr[j];
      tile[(k * TS + q) * 2 + 1] = ri[j];
    }
  }
  __syncthreads();
}

// ---------------- kernel 2: K_hat = fft2(pad(k)) half-spectrum ----------------
__global__ void __launch_bounds__(256) fftconv_khat_kernel(float* ws, const float* __restrict__ kin) {
  extern __shared__ float lds[];
  const int tid = threadIdx.x, lane = tid & 31, wave = tid >> 5;
  const int c = blockIdx.x;

  float* tile = lds + L_TILE;
  float* scrR = lds + L_SCR + wave * 544;
  float* scrI = scrR + 272;

  Ffrag f;
  load_ffrag(ws, lane, f);
  v8f twr = {}, twi = {};
  load_twiddles(ws, lane, twr, twi);

  forward_tile(kin + (size_t)c * (IH * IW), tile, scrR, scrI, f, twr, twi, tid, lane, wave);

  float* kh = ws + WS_KHAT + (size_t)c * (NP * NHW * 2);
  for (int i = tid; i < NP * NHW; i += 256) {
    int h = i / NHW, wq = i % NHW;
    kh[i * 2]     = tile[(h * TS + wq) * 2];
    kh[i * 2 + 1] = tile[(h * TS + wq) * 2 + 1];
  }
}

// ---------------- kernel 3: fused fwd-FFT * K_hat -> inv-FFT -> crop ----------------
__global__ void __launch_bounds__(256) fftconv_main_kernel(const float* __restrict__ ws,
                                                           const float* __restrict__ u,
                                                           float* __restrict__ out) {
  extern __shared__ float lds[];
  const int tid = threadIdx.x, lane = tid & 31, wave = tid >> 5;
  const int bc = blockIdx.x;               // 0..767
  const int c = bc % NC;

  float* tile = lds + L_TILE;
  float* scrR = lds + L_SCR + wave * 544;
  float* scrI = scrR + 272;

  Ffrag f;
  load_ffrag(ws, lane, f);
  v8f twr = {}, twi = {};
  load_twiddles(ws, lane, twr, twi);

  // forward: u tile -> half spectrum in LDS
  forward_tile(u + (size_t)bc * (IH * IW), tile, scrR, scrI, f, twr, twi, tid, lane, wave);

  // pointwise complex multiply with K_hat (k broadcasts over batch)
  const float* kh = ws + WS_KHAT + (size_t)c * (NP * NHW * 2);
  for (int i = tid; i < NP * NHW; i += 256) {
    __builtin_prefetch(kh + (i + 256) * 2, 0, 1);      // global_prefetch ahead of stream
    int h = i / NHW, wq = i % NHW;
    float ar = tile[(h * TS + wq) * 2], ai = tile[(h * TS + wq) * 2 + 1];
    float br = kh[i * 2], bi = kh[i * 2 + 1];
    tile[(h * TS + wq) * 2]     = ar * br - ai * bi;
    tile[(h * TS + wq) * 2 + 1] = ar * bi + ai * br;
  }
  __syncthreads();

  // inverse column FFTs: conj in, forward core, conj out (scale deferred)
  for (int q = wave; q < NHW; q += 8) {
#pragma unroll
    for (int e = 0; e < 8; ++e) {
      int n = lane * 8 + e;
      scrR[(n >> 4) * FS + (n & 15)] =  tile[(n * TS + q) * 2];
      scrI[(n >> 4) * FS + (n & 15)] = -tile[(n * TS + q) * 2 + 1];
    }
    v8f rr, ri;
    fft256_wave(scrR, scrI, f, twr, twi, lane, rr, ri);
    int col = lane & 15, rbase = (lane >> 4) * 8;
#pragma unroll
    for (int j = 0; j < 8; ++j) {
      int k = (rbase + j) + 16 * col;
      tile[(k * TS + q) * 2]     =  rr[j];
      tile[(k * TS + q) * 2 + 1] = -ri[j];
    }
  }
  __syncthreads();

  // inverse row FFTs with Hermitian reconstruction; conj folded into the fill,
  // output real part (unchanged by conj), scale 1/(256*256), crop to 224
  float* orow = out + (size_t)bc * (IH * IW);
  for (int r = wave; r < IH; r += 8) {
#pragma unroll
    for (int e = 0; e < 8; ++e) {
      int n = lane * 8 + e;
      float vr, vi;
      if (n < NHW) {            // conj(t[n])
        vr =  tile[(r * TS + n) * 2];
        vi = -tile[(r * TS + n) * 2 + 1];
      } else {                  // conj(conj(t[256-n])) = t[256-n]
        int m = NP - n;         // 1..127
        vr = tile[(r * TS + m) * 2];
        vi = tile[(r * TS + m) * 2 + 1];
      }
      scrR[(n >> 4) * FS + (n & 15)] = vr;
      scrI[(n >> 4) * FS + (n & 15)] = vi;
    }
    v8f rr, ri;
    fft256_wave(scrR, scrI, f, twr, twi, lane, rr, ri);
    int col = lane & 15, rbase = (lane >> 4) * 8;
    if (col < 14) {                                       // k = c_+16*col < 224
#pragma unroll
      for (int j = 0; j < 8; ++j) {
        int kx = (rbase + j) + 16 * col;
        orow[(size_t)r * IW + kx] = rr[j] * (1.0f / 65536.0f);
      }
    }
  }
}

extern "C" void kernel_launch(void* const* d_in, const int* in_sizes, int n_in,
                              void* d_out, int out_size, void* d_ws, size_t ws_size,
                              hipStream_t stream) {
  (void)in_sizes; (void)n_in; (void)out_size; (void)ws_size;
  const float* u = (const float*)d_in[0];
  const float* k = (const float*)d_in[1];
  float* out = (float*)d_out;
  float* ws = (float*)d_ws;

  const size_t ldsb = (size_t)LDS_FLOATS * sizeof(float);  // 283648 B (fits 320KB WGP LDS)

  fftconv_tables_kernel<<<1, 256, 0, stream>>>(ws);
  fftconv_khat_kernel<<<NC, 256, ldsb, stream>>>(ws, k);
  fftconv_main_kernel<<<NBATCH * NC, 256, ldsb, stream>>>(ws, u, out);
}